// Multi_shallow_embedding_11914239279603
// MI455X (gfx1250) — compile-verified
//
#include <hip/hip_runtime.h>

#define NNODES 1024
#define NGRAPH 128
#define TOPK   16384u
#define NTILE  64                          // 1024/16 tiles per dimension
#define TILES_PER_GRAPH (NTILE * NTILE)    // 4096

typedef __attribute__((ext_vector_type(2))) float v2f;
typedef __attribute__((ext_vector_type(4))) float v4f;
typedef __attribute__((ext_vector_type(8))) float v8f;

// Branchless monotone map: larger float <=> larger unsigned.
// Compiles to v_ashrrev_i32 + v_bitop3_b32 (2 VALU ops).
__device__ __forceinline__ unsigned mono(float f) {
    unsigned b = __float_as_uint(f);
    return b ^ ((unsigned)((int)b >> 31) | 0x80000000u);
}

// ---------------------------------------------------------------------------
// One radix pass over ALL 1M products s_i*t_j of one graph (diagonal included;
// corrected separately). Products come from V_WMMA_F32_16X16X4_F32 (rank-1:
// only K=0 nonzero -> exact f32 product). PASS is a template parameter so each
// pass is a specialized straight-line loop.
//   PASS 0: all elements, bin = u[31:21]   (2048 bins) -- atomic UNpredicated
//   PASS 1: u[31:21]==pfx, bin = u[20:10]  (2048 bins)
//   PASS 2: u[31:10]==pfx, bin = u[9:0]    (1024 bins)
// ---------------------------------------------------------------------------
template <int PASS>
__device__ __forceinline__ void histo_pass(const float* __restrict__ ls,
                                           const float* __restrict__ lt,
                                           unsigned* __restrict__ hist,
                                           unsigned pfx, int lane, int wave) {
    const bool lo  = (lane < 16);
    const int  l15 = lane & 15;

    for (int tIdx = wave; tIdx < TILES_PER_GRAPH; tIdx += 8) {
        const int ti = tIdx >> 6;
        const int tj = tIdx & 63;

        // Unconditional LDS reads (lanes 16-31 broadcast-read lanes 0-15's
        // addresses) + cndmask select: no exec-mask round trips.
        const float av = ls[ti * 16 + l15];
        const float bv = lt[tj * 16 + l15];

        // A 16x4 f32: lanes 0-15 hold M=0..15; VGPR0=K0 (lanes 0-15), K2 (16-31).
        // B 4x16 f32: row K striped across lanes within a VGPR, same split.
        v2f a, b;
        a.x = lo ? av : 0.0f;  a.y = 0.0f;
        b.x = lo ? bv : 0.0f;  b.y = 0.0f;

        v8f c = {};
        c = __builtin_amdgcn_wmma_f32_16x16x4_f32(
                /*neg_a=*/false, a, /*neg_b=*/false, b,
                /*c_mod=*/(short)0, c, /*reuse_a=*/false, /*reuse_b=*/false);

        #pragma unroll
        for (int r = 0; r < 8; ++r) {
            const unsigned u = mono(c[r]);
            if constexpr (PASS == 0) {
                atomicAdd(&hist[u >> 21], 1u);                 // no predication
            } else if constexpr (PASS == 1) {
                if ((u >> 21) == pfx) atomicAdd(&hist[(u >> 10) & 2047u], 1u);
            } else {
                if ((u >> 10) == pfx) atomicAdd(&hist[u & 1023u], 1u);
            }
        }
    }
}

// Subtract the 1024 diagonal products (reference masks them to -inf).
// v_mul_f32 is bit-identical to the rank-1 WMMA product (both RNE), so this
// exactly cancels their histogram contribution. Runs concurrently with the
// main pass (add/sub commute); no extra barrier needed.
template <int PASS>
__device__ __forceinline__ void diag_correct(const float* __restrict__ ls,
                                             const float* __restrict__ lt,
                                             unsigned* __restrict__ hist,
                                             unsigned pfx, int tid) {
    #pragma unroll
    for (int i = 0; i < NNODES / 256; ++i) {
        const int n = tid + i * 256;
        const unsigned u = mono(ls[n] * lt[n]);
        if constexpr (PASS == 0) {
            atomicSub(&hist[u >> 21], 1u);
        } else if constexpr (PASS == 1) {
            if ((u >> 21) == pfx) atomicSub(&hist[(u >> 10) & 2047u], 1u);
        } else {
            if ((u >> 10) == pfx) atomicSub(&hist[u & 1023u], 1u);
        }
    }
}

__device__ __forceinline__ void clear_hist(unsigned* hist, int tid) {
    #pragma unroll
    for (int i = 0; i < 8; ++i) hist[tid + i * 256] = 0u;
}

__device__ __forceinline__ void scan_hist(const unsigned* __restrict__ hist,
                                          unsigned* sh_pfx, unsigned* sh_rem,
                                          int nb, int bits, int tid) {
    if (tid == 0) {
        const unsigned rem = *sh_rem;
        unsigned cum = 0; int chosen = 0;
        for (int b = nb - 1; b >= 0; --b) {
            const unsigned c = hist[b];
            if (cum + c >= rem) { chosen = b; break; }
            cum += c;
        }
        *sh_rem = rem - cum;   // still needed inside the chosen bin (>= 1)
        *sh_pfx = (*sh_pfx << bits) | (unsigned)chosen;
    }
}

// ---------------------------------------------------------------------------
// Kernel 1: per-graph radix-select of the K-th largest off-diagonal product.
// One block (8 wave32) per graph; emits tau = bit pattern of K-th largest.
// ---------------------------------------------------------------------------
__global__ __launch_bounds__(256)
void topk_threshold_kernel(const float* __restrict__ emb_s,
                           const float* __restrict__ emb_t,
                           unsigned* __restrict__ tau_out) {
    const int g    = blockIdx.x;
    const int tid  = threadIdx.x;
    const int lane = tid & 31;
    const int wave = tid >> 5;

    __shared__ float    ls[NNODES];
    __shared__ float    lt[NNODES];
    __shared__ unsigned hist[2048];
    __shared__ unsigned sh_pfx;
    __shared__ unsigned sh_rem;

    #pragma unroll
    for (int i = 0; i < NNODES / 256; ++i) {
        ls[tid + i * 256] = emb_s[g * NNODES + tid + i * 256];
        lt[tid + i * 256] = emb_t[g * NNODES + tid + i * 256];
    }
    if (tid == 0) { sh_pfx = 0u; sh_rem = TOPK; }
    __syncthreads();

    // ---- pass 0 ----
    clear_hist(hist, tid);           __syncthreads();
    diag_correct<0>(ls, lt, hist, 0u, tid);
    histo_pass<0>(ls, lt, hist, 0u, lane, wave);
    __syncthreads();
    scan_hist(hist, &sh_pfx, &sh_rem, 2048, 11, tid);
    __syncthreads();
    unsigned pfx = sh_pfx;

    // ---- pass 1 ----
    clear_hist(hist, tid);           __syncthreads();
    diag_correct<1>(ls, lt, hist, pfx, tid);
    histo_pass<1>(ls, lt, hist, pfx, lane, wave);
    __syncthreads();
    scan_hist(hist, &sh_pfx, &sh_rem, 2048, 11, tid);
    __syncthreads();
    pfx = sh_pfx;

    // ---- pass 2 ----
    clear_hist(hist, tid);           __syncthreads();
    diag_correct<2>(ls, lt, hist, pfx, tid);
    histo_pass<2>(ls, lt, hist, pfx, lane, wave);
    __syncthreads();
    scan_hist(hist, &sh_pfx, &sh_rem, 1024, 10, tid);
    __syncthreads();

    if (tid == 0) tau_out[g] = sh_pfx;   // full 32-bit pattern (11+11+10)
}

// ---------------------------------------------------------------------------
// Kernel 2: stream the 0/1 mask. One thread per 4 consecutive columns;
// non-temporal B128 stores (512 MB written once, never re-read -> keep L2
// clean for emb_t which is re-read 1024x per graph).
// ---------------------------------------------------------------------------
__global__ __launch_bounds__(256)
void write_mask_kernel(const float* __restrict__ emb_s,
                       const float* __restrict__ emb_t,
                       const unsigned* __restrict__ tau_in,
                       float* __restrict__ out) {
    const unsigned idx = blockIdx.x * 256u + threadIdx.x;   // < G*N*N/4
    const unsigned g   = idx >> 18;                         // N*N/4 = 2^18
    const unsigned r   = idx & 0x3FFFFu;
    const unsigned i   = r >> 8;                            // row
    const unsigned j0  = (r & 255u) << 2;                   // first of 4 cols

    const float    sv = emb_s[g * NNODES + i];
    const unsigned tu = tau_in[g];

    const v4f tv = *reinterpret_cast<const v4f*>(&emb_t[g * NNODES + j0]);

    v4f o;
    o.x = (mono(sv * tv.x) >= tu && (j0 + 0u) != i) ? 1.0f : 0.0f;
    o.y = (mono(sv * tv.y) >= tu && (j0 + 1u) != i) ? 1.0f : 0.0f;
    o.z = (mono(sv * tv.z) >= tu && (j0 + 2u) != i) ? 1.0f : 0.0f;
    o.w = (mono(sv * tv.w) >= tu && (j0 + 3u) != i) ? 1.0f : 0.0f;

    __builtin_nontemporal_store(
        o, reinterpret_cast<v4f*>(&out[((size_t)g << 20) + ((size_t)i << 10) + j0]));
}

extern "C" void kernel_launch(void* const* d_in, const int* in_sizes, int n_in,
                              void* d_out, int out_size, void* d_ws, size_t ws_size,
                              hipStream_t stream) {
    (void)in_sizes; (void)n_in; (void)out_size; (void)ws_size;
    // inputs: d_in[0] = x (unused), d_in[1] = emb_s (G,N,1), d_in[2] = emb_t (G,1,N)
    const float* emb_s = (const float*)d_in[1];
    const float* emb_t = (const float*)d_in[2];
    float*    out = (float*)d_out;
    unsigned* tau = (unsigned*)d_ws;   // 128 * 4 bytes of scratch

    topk_threshold_kernel<<<NGRAPH, 256, 0, stream>>>(emb_s, emb_t, tau);

    const unsigned total4 = (unsigned)NGRAPH * NNODES * (NNODES / 4);  // 2^25
    write_mask_kernel<<<total4 / 256, 256, 0, stream>>>(emb_s, emb_t, tau, out);
}